// GenericNonlinearLocalLearnableTransitionLayer_89747636617230
// MI455X (gfx1250) — compile-verified
//
#include <hip/hip_runtime.h>

// ---------------------------------------------------------------------------
// Toroidal 3x3 depthwise box filter (uniform 1/27 weight) + per-pixel
// quadratic forms  delta_k = xhat^T M_k xhat + bias_k,  out = x + eps*delta.
//
// Bandwidth-bound: 201 MB total traffic -> ~8.6us floor @ 23.3 TB/s.
// Quadratic form on the CDNA5 matrix pipe via V_WMMA_F32_16X16X4_F32:
//   D(16x16) = A(16x4) x B(4x16) + C, pixels as B/D columns (= lanes).
//   A holds symmetrized coefficients Ssym[k][c] in rows 0..2 AND rows 8..10:
//     - rows 0..2  -> D rows 0..2  -> results for lanes 0..15  (pixel set A)
//     - rows 8..10 -> D rows 8..10 -> results for lanes 16..31 (pixel set B)
//   so each lane reads its own pixel's deltas from D vgprs 0..2 with a plain
//   cndmask — no cross-lane fold-back (avoids EXEC-predicated ds_swizzle).
//   B rows hold O' = {x0^2, x1^2, x2^2, x0*x1, x0*x2, x1*x2} (K=6 -> 2 WMMAs),
//   C holds bias in rows 0..2 and 8..10 (cbias[0..2] = bias in all lanes).
// ---------------------------------------------------------------------------

typedef float v2f __attribute__((ext_vector_type(2)));
typedef float v8f __attribute__((ext_vector_type(8)));

#define EPS_F 0.1f

// ds_swizzle_b32, group-of-32 mode: src_lane = ((lane & and) | or) ^ xor
// 0x000F: every lane reads lane&15        (broadcast lower half's pixels)
// 0x020F: every lane reads (lane&15)|16   (broadcast upper half's pixels)
// Both results are consumed by ALL lanes (they feed the WMMA B operands), so
// the compiler cannot predicate them.
template <int IMM>
__device__ __forceinline__ float dswz(float v) {
    return __builtin_bit_cast(
        float, __builtin_amdgcn_ds_swizzle(__builtin_bit_cast(int, v), IMM));
}

__device__ __forceinline__ void ld3(const float* __restrict__ p, float* d) {
    d[0] = p[0]; d[1] = p[1]; d[2] = p[2];   // adjacent -> global_load_b96
}

__global__ __launch_bounds__(256)
void gnllt_kernel(const float* __restrict__ x,
                  const float* __restrict__ Am,
                  const float* __restrict__ Bm,
                  const float* __restrict__ Cm,
                  const float* __restrict__ bias,
                  const float* __restrict__ w,
                  float* __restrict__ out) {
    const int H = 512, W = 512;
    const int lane = threadIdx.x & 31;
    const int wv   = threadIdx.x >> 5;

    const int row = blockIdx.x * 8 + wv;          // 0 .. 32*512-1
    const int b   = row >> 9;
    const int y   = row & 511;
    const int ym1 = (y + 511) & 511;              // toroidal wrap in H
    const int yp1 = (y + 1)   & 511;

    const float* r0 = x + (size_t)(b * H + ym1) * W * 3;
    const float* r1 = x + (size_t)(b * H + y  ) * W * 3;
    const float* r2 = x + (size_t)(b * H + yp1) * W * 3;
    float*      orow = out + (size_t)(b * H + y) * W * 3;

    const bool lo = lane < 16;
    const int  r  = lane & 15;                    // A-matrix row index

    // --- A matrix, 16x4 f32 layout: v.x = K0 | K2 across lane halves,
    //                                v.y = K1 | K3 across lane halves.
    // Coefficient rows 0..2 (for lane set A) and 8..10 (for lane set B).
    // Ssym[k] = {M00, M11, M22, M01+M10, M02+M20, M12+M21}
    const int k = (r < 3) ? r : ((r >= 8 && r < 11) ? (r - 8) : -1);
    v2f a1 = {0.f, 0.f};    // K chunk 0..3 -> Ssym[0..3]
    v2f a2 = {0.f, 0.f};    // K chunk 4..7 -> Ssym[4..5], rows 6,7 zero
    if (k >= 0) {
        const float* p = (k == 0) ? Am : (k == 1 ? Bm : Cm);
        const float s0 = p[0], s1 = p[4], s2 = p[8];
        const float s3 = p[1] + p[3];
        const float s4 = p[2] + p[6];
        const float s5 = p[5] + p[7];
        a1.x = lo ? s0 : s2;
        a1.y = lo ? s1 : s3;
        a2.x = lo ? s4 : 0.f;
        a2.y = lo ? s5 : 0.f;
    }

    // --- C matrix: bias in D rows 0..2 (lanes 0..15) and rows 8..10
    // (lanes 16..31) simultaneously: vgpr c holds bias[c] in every lane.
    v8f cbias = {};
    cbias[0] = bias[0];
    cbias[1] = bias[1];
    cbias[2] = bias[2];

    const float wgt = w[0];                       // uniform 1/(K*K*C) = 1/27

    const int colbase = lane * 16;                // lane's 16-column run

    // --- rolling column sums (3 input rows summed, per channel)
    float ca[3], cb[3], mid[3];
    {
        float t0[3], t1[3], t2[3];
        const int xm1 = (colbase - 1) & 511;      // toroidal wrap in W
        ld3(r0 + xm1 * 3, t0); ld3(r1 + xm1 * 3, t1); ld3(r2 + xm1 * 3, t2);
        for (int c = 0; c < 3; ++c) ca[c] = t0[c] + t1[c] + t2[c];
        ld3(r0 + colbase * 3, t0); ld3(r1 + colbase * 3, t1); ld3(r2 + colbase * 3, t2);
        for (int c = 0; c < 3; ++c) { cb[c] = t0[c] + t1[c] + t2[c]; mid[c] = t1[c]; }
    }

#pragma unroll 4
    for (int s = 0; s < 16; ++s) {
        const int xc = colbase + s;
        const int xp = (xc + 1) & 511;

        float t0[3], t1[3], t2[3], cc[3], midn[3];
        ld3(r0 + xp * 3, t0); ld3(r1 + xp * 3, t1); ld3(r2 + xp * 3, t2);
        for (int c = 0; c < 3; ++c) { cc[c] = t0[c] + t1[c] + t2[c]; midn[c] = t1[c]; }

        // xhat for this lane's pixel
        const float xh0 = (ca[0] + cb[0] + cc[0]) * wgt;
        const float xh1 = (ca[1] + cb[1] + cc[1]) * wgt;
        const float xh2 = (ca[2] + cb[2] + cc[2]) * wgt;

        // Broadcast each half-wave's pixels to all lanes (B columns = pixels).
        const float xA0 = dswz<0x000F>(xh0), xA1 = dswz<0x000F>(xh1), xA2 = dswz<0x000F>(xh2);
        const float xB0 = dswz<0x020F>(xh0), xB1 = dswz<0x020F>(xh1), xB2 = dswz<0x020F>(xh2);

        // B rows (4x16 layout: v.x = K0|K2, v.y = K1|K3 across lane halves):
        //   chunk0: K0=x0^2, K1=x1^2, K2=x2^2, K3=x0*x1
        //   chunk1: K4=x0*x2, K5=x1*x2, K6=K7=0
        v2f b1A, b2A, b1B, b2B;
        b1A.x = (lo ? xA0 : xA2) * (lo ? xA0 : xA2);
        b1A.y = (lo ? xA1 : xA0) * xA1;
        b2A.x = (lo ? xA0 : 0.f) * xA2;
        b2A.y = (lo ? xA1 : 0.f) * xA2;
        b1B.x = (lo ? xB0 : xB2) * (lo ? xB0 : xB2);
        b1B.y = (lo ? xB1 : xB0) * xB1;
        b2B.x = (lo ? xB0 : 0.f) * xB2;
        b2B.y = (lo ? xB1 : 0.f) * xB2;

        // D = A*B(chunk0) accumulated into A*B(chunk1) + bias, per pixel set.
        v8f dA = __builtin_amdgcn_wmma_f32_16x16x4_f32(false, a2, false, b2A,
                                                       (short)0, cbias, false, false);
        dA     = __builtin_amdgcn_wmma_f32_16x16x4_f32(false, a1, false, b1A,
                                                       (short)0, dA, false, false);
        v8f dB = __builtin_amdgcn_wmma_f32_16x16x4_f32(false, a2, false, b2B,
                                                       (short)0, cbias, false, false);
        dB     = __builtin_amdgcn_wmma_f32_16x16x4_f32(false, a1, false, b1B,
                                                       (short)0, dB, false, false);

        // Lane's own deltas: lanes 0..15 from dA rows 0..2, lanes 16..31 from
        // dB rows 8..10 (which live in vgprs 0..2 for the upper half). Plain
        // per-lane selects — no cross-lane ops, no EXEC games.
        const float d0 = lo ? dA[0] : dB[0];
        const float d1 = lo ? dA[1] : dB[1];
        const float d2 = lo ? dA[2] : dB[2];

        float* op = orow + (size_t)xc * 3;        // adjacent -> global_store_b96
        op[0] = fmaf(EPS_F, d0, mid[0]);
        op[1] = fmaf(EPS_F, d1, mid[1]);
        op[2] = fmaf(EPS_F, d2, mid[2]);

        for (int c = 0; c < 3; ++c) { ca[c] = cb[c]; cb[c] = cc[c]; mid[c] = midn[c]; }
    }
}

extern "C" void kernel_launch(void* const* d_in, const int* in_sizes, int n_in,
                              void* d_out, int out_size, void* d_ws, size_t ws_size,
                              hipStream_t stream) {
    (void)in_sizes; (void)n_in; (void)out_size; (void)d_ws; (void)ws_size;
    const float* x    = (const float*)d_in[0];
    const float* a    = (const float*)d_in[1];
    const float* b    = (const float*)d_in[2];
    const float* c    = (const float*)d_in[3];
    const float* bias = (const float*)d_in[4];
    const float* w    = (const float*)d_in[5];
    float* out        = (float*)d_out;

    // 32*512 rows / (8 waves per block) = 2048 blocks of 256 threads.
    gnllt_kernel<<<2048, 256, 0, stream>>>(x, a, b, c, bias, w, out);
}